// RoadCaps_12747462934975
// MI455X (gfx1250) — compile-verified
//
#include <hip/hip_runtime.h>
#include <hip/hip_bf16.h>
#include <math.h>
#include <stdint.h>

typedef float v2f __attribute__((ext_vector_type(2)));
typedef float v8f __attribute__((ext_vector_type(8)));
typedef unsigned int v4u __attribute__((ext_vector_type(4)));
typedef int v8i __attribute__((ext_vector_type(8)));
typedef int v4i __attribute__((ext_vector_type(4)));

#define N_NODES   100000
#define N_EDGES   1600000
#define N_FEAT    128
#define GCN_F     64
#define CAP_D     8
#define N_CAPS    8
#define N_TGT     6
#define N_B       12500   // N_NODES*CAP_D/(CAP_D*N_CAPS)

// ---------------------------------------------------------------------------
// Degree / normalization
// ---------------------------------------------------------------------------
__global__ __launch_bounds__(256) void k_deg_init(float* deg) {
  int n = blockIdx.x * 256 + threadIdx.x;
  if (n < N_NODES) deg[n] = 1.0f;               // self loop
}
__global__ __launch_bounds__(256) void k_deg_count(const int* __restrict__ dst, float* deg) {
  int e = blockIdx.x * 256 + threadIdx.x;
  if (e < N_EDGES) atomicAdd(&deg[dst[e]], 1.0f);
}
__global__ __launch_bounds__(256) void k_deg_rsqrt(float* deg) {
  int n = blockIdx.x * 256 + threadIdx.x;
  if (n < N_NODES) deg[n] = rsqrtf(fmaxf(deg[n], 1e-12f));
}
__global__ __launch_bounds__(256) void k_zero_scalars(float* sc) {
  sc[threadIdx.x] = 0.0f;                       // 256 slots
}

// ---------------------------------------------------------------------------
// WMMA GEMM: C[M,64] = A[M,K] @ B[K,64]   (row-major, f32, K in {64,128})
// 8 waves/block, one 16x64 tile per wave.
// B staged into LDS by the Tensor Data Mover (one TDM DMA per block,
// issued by wave 0, drained with s_wait_tensorcnt, published by barrier).
// ---------------------------------------------------------------------------
__global__ __launch_bounds__(256) void k_gemm_wmma(const float* __restrict__ A,
                                                   const float* __restrict__ Bw,
                                                   float* __restrict__ C,
                                                   int M, int K) {
  __shared__ float sB[N_FEAT * 64];             // up to 32 KB

  if (threadIdx.x < 32) {                       // wave 0 drives the TDM
    unsigned long long ga = (unsigned long long)(uintptr_t)Bw;
    unsigned lds = (unsigned)(uintptr_t)&sB[0]; // LDS byte offset (addrspace 3)
    // D# group 0: count=1 | lds_addr | global_addr[56:0] | type=2
    v4u g0 = { 1u,
               lds,
               (unsigned)(ga & 0xFFFFFFFFu),
               (unsigned)((ga >> 32) & 0x01FFFFFFu) | 0x80000000u };
    // D# group 1: data_size=4B; tensor 64 x K (stride 64); tile 64 x K
    v8i g1 = { (int)(2u << 16),                 // data_size=2 -> 4 bytes
               (int)(64u << 16),                // tensor_dim0 = 64  (bits 79:48 lo)
               (int)((unsigned)K << 16),        // tensor_dim0 hi=0 | tensor_dim1 = K (lo16)
               (int)(64u << 16),                // tensor_dim1 hi=0 | tile_dim0 = 64
               K,                               // tile_dim1 = K, tile_dim2 = 0
               64,                              // tensor_dim0_stride = 64 (lo32)
               0,                               // stride hi | tensor_dim1_stride lo
               0 };
    v4i g2 = { 0, 0, 0, 0 };                    // 2D tensor: higher dims unused
    v4i g3 = { 0, 0, 0, 0 };
#if defined(__clang_major__) && (__clang_major__ >= 23)
    v8i g4 = { 0, 0, 0, 0, 0, 0, 0, 0 };
    __builtin_amdgcn_tensor_load_to_lds(g0, g1, g2, g3, g4, 0);
#else
    __builtin_amdgcn_tensor_load_to_lds(g0, g1, g2, g3, 0);
#endif
    __builtin_amdgcn_s_wait_tensorcnt((short)0);
  }
  __syncthreads();

  const int wave = threadIdx.x >> 5;
  const int lane = threadIdx.x & 31;
  const int tile = blockIdx.x * 8 + wave;
  if (tile * 16 >= M) return;

  const int n   = lane & 15;
  const int kg  = (lane >> 4) << 1;             // 0 or 2
  const int row = tile * 16 + (lane & 15);

  v8f acc0 = {}, acc1 = {}, acc2 = {}, acc3 = {};
  for (int k = 0; k < K; k += 4) {
    v2f a;
    a.x = A[row * K + k + kg];
    a.y = A[row * K + k + kg + 1];
    const float* b0p = &sB[(k + kg) * 64];
    const float* b1p = &sB[(k + kg + 1) * 64];
    v2f b;
    b.x = b0p[n +  0]; b.y = b1p[n +  0];
    acc0 = __builtin_amdgcn_wmma_f32_16x16x4_f32(false, a, false, b, (short)0, acc0, false, false);
    b.x = b0p[n + 16]; b.y = b1p[n + 16];
    acc1 = __builtin_amdgcn_wmma_f32_16x16x4_f32(false, a, false, b, (short)0, acc1, false, false);
    b.x = b0p[n + 32]; b.y = b1p[n + 32];
    acc2 = __builtin_amdgcn_wmma_f32_16x16x4_f32(false, a, false, b, (short)0, acc2, false, false);
    b.x = b0p[n + 48]; b.y = b1p[n + 48];
    acc3 = __builtin_amdgcn_wmma_f32_16x16x4_f32(false, a, false, b, (short)0, acc3, false, false);
  }
  const int mbase = tile * 16 + ((lane >> 4) << 3);
#pragma unroll
  for (int r = 0; r < 8; ++r) {
    float* o = &C[(mbase + r) * 64 + n];
    o[ 0] = acc0[r];
    o[16] = acc1[r];
    o[32] = acc2[r];
    o[48] = acc3[r];
  }
}

// ---------------------------------------------------------------------------
// GCN aggregation:  h[d] = sum_edges dinv[s]*dinv[d]*xw[s]  + dinv[n]^2*xw[n]
// then h = relu(h + b)
// ---------------------------------------------------------------------------
__global__ __launch_bounds__(256) void k_agg_self(const float* __restrict__ xw,
                                                  const float* __restrict__ dinv,
                                                  float* __restrict__ h) {
  int idx = blockIdx.x * 256 + threadIdx.x;     // N_NODES*64
  if (idx < N_NODES * 64) {
    int n = idx >> 6;
    float di = dinv[n];
    h[idx] = xw[idx] * di * di;
  }
}
__global__ __launch_bounds__(256) void k_agg_edges(const float* __restrict__ xw,
                                                   const int* __restrict__ src,
                                                   const int* __restrict__ dst,
                                                   const float* __restrict__ dinv,
                                                   float* __restrict__ h) {
  int idx = blockIdx.x * 256 + threadIdx.x;     // N_EDGES*16 (4 feats each)
  if (idx >= N_EDGES * 16) return;
  int e  = idx >> 4;
  int f4 = (idx & 15) << 2;
  int s = src[e], d = dst[e];
  float w = dinv[s] * dinv[d];
  const float4 v = *(const float4*)&xw[s * 64 + f4];
  float* o = &h[d * 64 + f4];
  atomicAdd(o + 0, w * v.x);
  atomicAdd(o + 1, w * v.y);
  atomicAdd(o + 2, w * v.z);
  atomicAdd(o + 3, w * v.w);
}
__global__ __launch_bounds__(256) void k_bias_relu(float* __restrict__ h,
                                                   const float* __restrict__ b) {
  int idx = blockIdx.x * 256 + threadIdx.x;
  if (idx < N_NODES * 64) h[idx] = fmaxf(h[idx] + b[idx & 63], 0.0f);
}

// ---------------------------------------------------------------------------
// Primary capsules: u[d,n] = sum_f Wp[d,f] * h_flat[f*N + n] + bp[d]
// plus global sum of squares (for the whole-vector squash).
// ---------------------------------------------------------------------------
__global__ __launch_bounds__(256) void k_primary_caps(const float* __restrict__ hflat,
                                                      const float* __restrict__ Wp,
                                                      const float* __restrict__ bp,
                                                      float* __restrict__ u,
                                                      float* __restrict__ magsq) {
  __shared__ float red[256];
  int n = blockIdx.x * 256 + threadIdx.x;
  float local = 0.0f;
  if (n < N_NODES) {
    float hv[GCN_F];
#pragma unroll
    for (int f = 0; f < GCN_F; ++f) hv[f] = hflat[f * N_NODES + n];
#pragma unroll
    for (int d = 0; d < CAP_D; ++d) {
      float acc = bp[d];
#pragma unroll
      for (int f = 0; f < GCN_F; ++f) acc += Wp[d * GCN_F + f] * hv[f];
      u[d * N_NODES + n] = acc;
      local += acc * acc;
    }
  }
  red[threadIdx.x] = local;
  __syncthreads();
  for (int off = 128; off > 0; off >>= 1) {
    if (threadIdx.x < off) red[threadIdx.x] += red[threadIdx.x + off];
    __syncthreads();
  }
  if (threadIdx.x == 0) atomicAdd(magsq, red[0]);
}

// ---------------------------------------------------------------------------
// u_hat[b,c,t,i] = sum_j Wcap[c,t,i,j] * (uflat[b*64+j*8+c] * squash_scale)
// ---------------------------------------------------------------------------
__global__ __launch_bounds__(256) void k_uhat(const float* __restrict__ uflat,
                                              const float* __restrict__ Wcap,
                                              const float* __restrict__ magsq,
                                              float* __restrict__ uhat) {
  int idx = blockIdx.x * 256 + threadIdx.x;     // B*C*T = 600000
  if (idx >= N_B * N_CAPS * N_TGT) return;
  int t = idx % N_TGT;
  int c = (idx / N_TGT) % N_CAPS;
  int b = idx / (N_TGT * N_CAPS);
  float S = *magsq;
  float scale = sqrtf(S) / (1.0f + S);          // (S/(1+S))/sqrt(S)
  float x[CAP_D];
#pragma unroll
  for (int j = 0; j < CAP_D; ++j) x[j] = uflat[b * 64 + j * 8 + c] * scale;
  const float* W = &Wcap[(c * N_TGT + t) * CAP_D * CAP_D];
  float* o = &uhat[((b * N_CAPS + c) * N_TGT + t) * CAP_D];
#pragma unroll
  for (int i = 0; i < CAP_D; ++i) {
    float acc = 0.0f;
#pragma unroll
    for (int j = 0; j < CAP_D; ++j) acc += W[i * CAP_D + j] * x[j];
    o[i] = acc;
  }
}

// ---------------------------------------------------------------------------
// One routing iteration: softmax(b_ij over T) -> s_j -> squash over T ->
// agreement delta[c,t] (atomic), optionally accumulate v for final mean.
// ---------------------------------------------------------------------------
__global__ __launch_bounds__(256) void k_route_iter(const float* __restrict__ uhat,
                                                    const float* __restrict__ bij,
                                                    float* __restrict__ delta,
                                                    float* __restrict__ vacc,
                                                    int accum_v) {
  __shared__ float cs[N_CAPS * N_TGT];
  __shared__ float red[256];
  __shared__ float sv[256 * 48];                // per-thread s/v values
  int tid = threadIdx.x;
  if (tid < N_CAPS) {
    int c = tid;
    float m = -1e30f;
    for (int t = 0; t < N_TGT; ++t) m = fmaxf(m, bij[c * N_TGT + t]);
    float e[N_TGT], ssum = 0.0f;
    for (int t = 0; t < N_TGT; ++t) { e[t] = expf(bij[c * N_TGT + t] - m); ssum += e[t]; }
    for (int t = 0; t < N_TGT; ++t) cs[c * N_TGT + t] = e[t] / ssum;
  }
  __syncthreads();

  int b = blockIdx.x * 256 + tid;
  bool act = b < N_B;
  float* my = &sv[tid * 48];
  for (int k = 0; k < 48; ++k) my[k] = 0.0f;

  if (act) {
    for (int c = 0; c < N_CAPS; ++c) {
      const float* up = &uhat[(b * N_CAPS + c) * N_TGT * CAP_D];
      for (int t = 0; t < N_TGT; ++t) {
        float cc = cs[c * N_TGT + t];
#pragma unroll
        for (int i = 0; i < CAP_D; ++i) my[t * 8 + i] += cc * up[t * 8 + i];
      }
    }
    // squash over T axis for each i
    for (int i = 0; i < CAP_D; ++i) {
      float ms = 0.0f;
      for (int t = 0; t < N_TGT; ++t) { float x = my[t * 8 + i]; ms += x * x; }
      float sc2 = sqrtf(ms) / (1.0f + ms);
      for (int t = 0; t < N_TGT; ++t) my[t * 8 + i] *= sc2;
    }
  }

  // agreement: delta[c,t] += sum_{b,i} uhat[b,c,t,i] * v[b,t,i]
  for (int c = 0; c < N_CAPS; ++c) {
    for (int t = 0; t < N_TGT; ++t) {
      float a = 0.0f;
      if (act) {
        const float* up = &uhat[((b * N_CAPS + c) * N_TGT + t) * CAP_D];
#pragma unroll
        for (int i = 0; i < CAP_D; ++i) a += up[i] * my[t * 8 + i];
      }
      red[tid] = a;
      __syncthreads();
      for (int off = 128; off > 0; off >>= 1) {
        if (tid < off) red[tid] += red[tid + off];
        __syncthreads();
      }
      if (tid == 0) atomicAdd(&delta[c * N_TGT + t], red[0]);
      __syncthreads();
    }
  }
  if (accum_v) {
    for (int k = 0; k < 48; ++k) {
      red[tid] = act ? my[k] : 0.0f;
      __syncthreads();
      for (int off = 128; off > 0; off >>= 1) {
        if (tid < off) red[tid] += red[tid + off];
        __syncthreads();
      }
      if (tid == 0) atomicAdd(&vacc[k], red[0]);
      __syncthreads();
    }
  }
}

__global__ void k_route_apply(float* bij, float* delta) {
  int k = threadIdx.x;
  if (k < 48) {
    bij[k] += delta[k] * (1.0f / (float)N_B);
    delta[k] = 0.0f;
  }
}

__global__ void k_final(const float* __restrict__ vacc, float* __restrict__ out) {
  int k = threadIdx.x;
  if (k < 48) out[k] = vacc[k] * (1.0f / (float)N_B);
}

// ---------------------------------------------------------------------------
// Launch
// ---------------------------------------------------------------------------
extern "C" void kernel_launch(void* const* d_in, const int* in_sizes, int n_in,
                              void* d_out, int out_size, void* d_ws, size_t ws_size,
                              hipStream_t stream) {
  const float* features = (const float*)d_in[0];
  const float* W0  = (const float*)d_in[1];
  const float* b0  = (const float*)d_in[2];
  const float* W1  = (const float*)d_in[3];
  const float* b1  = (const float*)d_in[4];
  const float* W2  = (const float*)d_in[5];
  const float* b2  = (const float*)d_in[6];
  const float* Wp  = (const float*)d_in[7];
  const float* bp  = (const float*)d_in[8];
  const float* Wcap= (const float*)d_in[9];
  const int*   edges = (const int*)d_in[10];
  const int* src = edges;
  const int* dst = edges + N_EDGES;

  float* ws = (float*)d_ws;
  float* dinv = ws;                              // 100000
  float* xw   = ws + 102400;                     // 6,400,000
  float* h    = xw + 6400000;                    // 6,400,000
  float* u    = h  + 6400000;                    // 800,000
  float* uhat = u  + 800000;                     // 4,800,000
  float* sc   = uhat + 4800000;                  // 256 scalar slots
  float* magsq = sc;                             // [0]
  float* bij   = sc + 16;                        // 48
  float* delta = sc + 64;                        // 48
  float* vacc  = sc + 128;                       // 48

  const int NB_NODE   = (N_NODES + 255) / 256;
  const int NB_EDGE   = (N_EDGES + 255) / 256;
  const int NB_NF     = (N_NODES * 64 + 255) / 256;          // 25000
  const int NB_EF     = (N_EDGES * 16 + 255) / 256;          // 100000
  const int NB_GEMM   = (N_NODES / 16 + 7) / 8;              // 782
  const int NB_UHAT   = (N_B * N_CAPS * N_TGT + 255) / 256;  // 2344
  const int NB_ROUTE  = (N_B + 255) / 256;                   // 49

  // normalization + state init
  k_deg_init    <<<NB_NODE, 256, 0, stream>>>(dinv);
  k_deg_count   <<<NB_EDGE, 256, 0, stream>>>(dst, dinv);
  k_deg_rsqrt   <<<NB_NODE, 256, 0, stream>>>(dinv);
  k_zero_scalars<<<1, 256, 0, stream>>>(sc);

  // GCN layer 0 (K=128)
  k_gemm_wmma <<<NB_GEMM, 256, 0, stream>>>(features, W0, xw, N_NODES, N_FEAT);
  k_agg_self  <<<NB_NF, 256, 0, stream>>>(xw, dinv, h);
  k_agg_edges <<<NB_EF, 256, 0, stream>>>(xw, src, dst, dinv, h);
  k_bias_relu <<<NB_NF, 256, 0, stream>>>(h, b0);
  // GCN layer 1 (K=64)
  k_gemm_wmma <<<NB_GEMM, 256, 0, stream>>>(h, W1, xw, N_NODES, GCN_F);
  k_agg_self  <<<NB_NF, 256, 0, stream>>>(xw, dinv, h);
  k_agg_edges <<<NB_EF, 256, 0, stream>>>(xw, src, dst, dinv, h);
  k_bias_relu <<<NB_NF, 256, 0, stream>>>(h, b1);
  // GCN layer 2 (K=64)
  k_gemm_wmma <<<NB_GEMM, 256, 0, stream>>>(h, W2, xw, N_NODES, GCN_F);
  k_agg_self  <<<NB_NF, 256, 0, stream>>>(xw, dinv, h);
  k_agg_edges <<<NB_EF, 256, 0, stream>>>(xw, src, dst, dinv, h);
  k_bias_relu <<<NB_NF, 256, 0, stream>>>(h, b2);

  // primary capsules + global squash magnitude
  k_primary_caps<<<NB_NODE, 256, 0, stream>>>(h, Wp, bp, u, magsq);
  k_uhat        <<<NB_UHAT, 256, 0, stream>>>(u, Wcap, magsq, uhat);

  // dynamic routing (3 iterations; last one accumulates v for the mean)
  k_route_iter <<<NB_ROUTE, 256, 0, stream>>>(uhat, bij, delta, vacc, 0);
  k_route_apply<<<1, 64, 0, stream>>>(bij, delta);
  k_route_iter <<<NB_ROUTE, 256, 0, stream>>>(uhat, bij, delta, vacc, 0);
  k_route_apply<<<1, 64, 0, stream>>>(bij, delta);
  k_route_iter <<<NB_ROUTE, 256, 0, stream>>>(uhat, bij, delta, vacc, 1);

  k_final<<<1, 64, 0, stream>>>(vacc, (float*)d_out);
}